// GraphAttentionHead_18545668784269
// MI455X (gfx1250) — compile-verified
//
#include <hip/hip_runtime.h>

typedef __attribute__((ext_vector_type(2))) float v2f;
typedef __attribute__((ext_vector_type(4))) float v4f;
typedef __attribute__((ext_vector_type(8))) float v8f;

#define BB 2
#define NN 768
#define EN 64
#define EE 16
#define FF 64

// Workspace layout (float offsets)
#define H_OFF   0                       // h: B*N*F = 98304
#define SR_OFF  (BB * NN * FF)          // sr: B*N = 1536
#define SC_OFF  (SR_OFF + BB * NN)      // sc: B*N = 1536
#define VE_OFF  (SC_OFF + BB * NN)      // v_edge: 16 (padded to 64)
#define ATT_OFF (VE_OFF + 64)           // att: B*N*N = 1179648  (even -> 8B aligned)

// ---------------------------------------------------------------------------
// Kernel A: h = nodes @ W_node  (fp32 WMMA, one 16x16 tile per wave, K=64)
// ---------------------------------------------------------------------------
__global__ void gat_h_wmma(const float* __restrict__ nodes,
                           const float* __restrict__ Wn,
                           float* __restrict__ ws) {
  float* __restrict__ h = ws + H_OFF;
  const int tile = blockIdx.x;                       // B * 48 * 4 tiles
  const int b    = tile / ((NN / 16) * (FF / 16));
  const int rem  = tile % ((NN / 16) * (FF / 16));
  const int m0   = (rem / (FF / 16)) * 16;
  const int n0   = (rem % (FF / 16)) * 16;
  const int lane = threadIdx.x & 31;
  const int half = lane >> 4;                        // K-pair selector
  const int ln   = lane & 15;                        // M (for A) / N (for B)

  const float* arow = nodes + (size_t)(b * NN + m0 + ln) * EN;
  v8f acc = {};
  #pragma unroll 4
  for (int k = 0; k < EN; k += 4) {
    const int ka = k + half * 2;
    v2f a = *(const v2f*)(arow + ka);                // A[M=ln][ka, ka+1]
    v2f bv;
    bv.x = Wn[(size_t)ka       * FF + n0 + ln];      // B[ka  ][N=ln]
    bv.y = Wn[(size_t)(ka + 1) * FF + n0 + ln];      // B[ka+1][N=ln]
    acc = __builtin_amdgcn_wmma_f32_16x16x4_f32(
        false, a, false, bv, (short)0, acc, false, false);
  }
  // C/D layout: VGPR v, lanes 0-15 -> M=v, lanes 16-31 -> M=v+8; N=ln
  float* ho = h + (size_t)(b * NN + m0 + half * 8) * FF + n0 + ln;
  #pragma unroll
  for (int v = 0; v < 8; ++v) ho[(size_t)v * FF] = acc[v];
}

// ---------------------------------------------------------------------------
// Kernel B: sr = h@a_row, sc = h@a_col, v_edge = W_edge@a_edge (tiny)
// ---------------------------------------------------------------------------
__global__ void gat_prep(const float* __restrict__ Wedge,
                         const float* __restrict__ a_w,
                         float* __restrict__ ws) {
  const float* h  = ws + H_OFF;
  float* sr = ws + SR_OFF;
  float* sc = ws + SC_OFF;
  float* ve = ws + VE_OFF;
  const int idx = blockIdx.x * blockDim.x + threadIdx.x;
  if (idx < BB * NN) {
    const float* hr = h + (size_t)idx * FF;
    float r = 0.f, c = 0.f;
    for (int f = 0; f < FF; ++f) {
      const float hv = hr[f];
      r += hv * a_w[f];              // a_row = a_w[0:64]
      c += hv * a_w[2 * FF + f];     // a_col = a_w[128:192]
    }
    sr[idx] = r;
    sc[idx] = c;
  }
  if (idx < EE) {
    float v = 0.f;
    for (int f = 0; f < FF; ++f) v += Wedge[idx * FF + f] * a_w[FF + f];
    ve[idx] = v;                     // a_edge = a_w[64:128]
  }
}

// ---------------------------------------------------------------------------
// Kernel C: scores + masked softmax per row (b,i).  Streams edges once (b128).
// ---------------------------------------------------------------------------
__global__ void gat_scores(const float* __restrict__ edges,
                           const unsigned char* __restrict__ mask,
                           float* __restrict__ ws) {
  __shared__ float ve_s[EE];
  __shared__ float red[256];
  const float* sr  = ws + SR_OFF;
  const float* sc  = ws + SC_OFF;
  const float* ve  = ws + VE_OFF;
  float*       att = ws + ATT_OFF;

  const int row = blockIdx.x;            // b*N + i
  const int b   = row / NN;
  const int tid = threadIdx.x;

  if (tid < EE) ve_s[tid] = ve[tid];
  __syncthreads();

  const float sri = sr[row];
  const float mi  = mask[row] ? 1.f : 0.f;
  const float* erow = edges + (size_t)row * NN * EE;

  float aw[3];
  #pragma unroll
  for (int t = 0; t < 3; ++t) {
    const int j = tid + t * 256;
    const v4f* e4 = (const v4f*)(erow + (size_t)j * EE);  // 64B aligned
    float d = 0.f;
    #pragma unroll
    for (int q = 0; q < 4; ++q) {
      const v4f e = e4[q];
      d += e.x * ve_s[q * 4 + 0] + e.y * ve_s[q * 4 + 1] +
           e.z * ve_s[q * 4 + 2] + e.w * ve_s[q * 4 + 3];
    }
    const float s = sri + sc[b * NN + j] + d;
    const float v = (s >= 0.f) ? s : 0.2f * s;             // leaky_relu 0.2
    aw[t] = v * mi;                                        // row mask on i
  }

  // row max
  red[tid] = fmaxf(aw[0], fmaxf(aw[1], aw[2]));
  __syncthreads();
  for (int off = 128; off > 0; off >>= 1) {
    if (tid < off) red[tid] = fmaxf(red[tid], red[tid + off]);
    __syncthreads();
  }
  const float rmax = red[0];
  __syncthreads();

  // exp + sum
  float p[3], lsum = 0.f;
  #pragma unroll
  for (int t = 0; t < 3; ++t) { p[t] = __expf(aw[t] - rmax); lsum += p[t]; }
  red[tid] = lsum;
  __syncthreads();
  for (int off = 128; off > 0; off >>= 1) {
    if (tid < off) red[tid] += red[tid + off];
    __syncthreads();
  }
  const float inv = 1.f / red[0];

  // att = softmax * (aw > 0)   (sigmoid(aw)>0.5  <=>  aw>0)
  #pragma unroll
  for (int t = 0; t < 3; ++t) {
    const int j = tid + t * 256;
    att[(size_t)row * NN + j] = (aw[t] > 0.f) ? p[t] * inv : 0.f;
  }
}

// ---------------------------------------------------------------------------
// Kernel D: out = att @ h  (fp32 WMMA, one 16x16 tile per wave, K=768)
// ---------------------------------------------------------------------------
__global__ void gat_out_wmma(float* __restrict__ out,
                             const float* __restrict__ ws) {
  const float* h   = ws + H_OFF;
  const float* att = ws + ATT_OFF;
  const int tile = blockIdx.x;
  const int b    = tile / ((NN / 16) * (FF / 16));
  const int rem  = tile % ((NN / 16) * (FF / 16));
  const int m0   = (rem / (FF / 16)) * 16;
  const int n0   = (rem % (FF / 16)) * 16;
  const int lane = threadIdx.x & 31;
  const int half = lane >> 4;
  const int ln   = lane & 15;

  const float* arow = att + (size_t)(b * NN + m0 + ln) * NN;
  const float* hb   = h + (size_t)b * NN * FF + n0 + ln;
  v8f acc = {};
  #pragma unroll 4
  for (int k = 0; k < NN; k += 4) {
    const int ka = k + half * 2;
    v2f a = *(const v2f*)(arow + ka);
    v2f bv;
    bv.x = hb[(size_t)ka       * FF];
    bv.y = hb[(size_t)(ka + 1) * FF];
    acc = __builtin_amdgcn_wmma_f32_16x16x4_f32(
        false, a, false, bv, (short)0, acc, false, false);
  }
  float* o = out + (size_t)(b * NN + m0 + half * 8) * FF + n0 + ln;
  #pragma unroll
  for (int v = 0; v < 8; ++v) o[(size_t)v * FF] = acc[v];
}

// ---------------------------------------------------------------------------
extern "C" void kernel_launch(void* const* d_in, const int* in_sizes, int n_in,
                              void* d_out, int out_size, void* d_ws, size_t ws_size,
                              hipStream_t stream) {
  (void)in_sizes; (void)n_in; (void)out_size; (void)ws_size;
  const float*         nodes = (const float*)d_in[0];
  const float*         edges = (const float*)d_in[1];
  const unsigned char* mask  = (const unsigned char*)d_in[2];  // bool (1B)
  const float*         Wn    = (const float*)d_in[3];
  const float*         We    = (const float*)d_in[4];
  const float*         a_w   = (const float*)d_in[5];
  float* out = (float*)d_out;
  float* ws  = (float*)d_ws;

  const int tiles = BB * (NN / 16) * (FF / 16);   // 384 waves
  gat_h_wmma  <<<tiles, 32, 0, stream>>>(nodes, Wn, ws);
  gat_prep    <<<(BB * NN + 255) / 256, 256, 0, stream>>>(We, a_w, ws);
  gat_scores  <<<BB * NN, 256, 0, stream>>>(edges, mask, ws);
  gat_out_wmma<<<tiles, 32, 0, stream>>>(out, ws);
}